// Mamba2_70738111365345
// MI455X (gfx1250) — compile-verified
//
#include <hip/hip_runtime.h>
#include <hip/hip_bf16.h>
#include <math.h>

// ---------------------------------------------------------------------------
// Mamba2 forward for MI455X (gfx1250, wave32, WMMA).
// All contractions use V_WMMA_F32_16X16X4_F32 (f32 in/out, matches reference
// numerics). Layout convention (from CDNA5 ISA 7.12.2, 32-bit 16x4 A-matrix):
//   A-frag (v2f): lane l holds M = l%16, K = k0 + (l/16)*2 + {0,1}
//   B-frag (v2f): lane l holds N = l%16, K = k0 + (l/16)*2 + {0,1}
//   C/D  (v8f):  element e: M = e + (l>=16 ? 8 : 0), N = l%16
// NT GEMMs (both operands K-contiguous) load both fragments as 8-byte pairs.
// All hot-loop addressing uses 32-bit element offsets (buffers < 2GB) so the
// compiler emits SGPR-base + VGPR32-offset global loads; the K loop is
// software-pipelined (double-buffered fragments) to overlap loads with WMMA.
// Workspace requirement: ~340 MB (see OFF_* below).
// ---------------------------------------------------------------------------

#define BATCH   2
#define SEQLEN  4096
#define DMODEL  1024
#define DSTATE  128
#define DCONV   4
#define DINNER  2048
#define DSSM    2048
#define NHEADS  32
#define HEADDIM 64
#define CHUNK   256
#define NCHUNK  (SEQLEN / CHUNK)                 // 16
#define CONVDIM (DSSM + 2 * DSTATE)              // 2304
#define DINPROJ (2 * DINNER + 2 * DSTATE + NHEADS) // 4384
#define DINPROJ_PAD 4480                         // ceil(4384/128)*128
#define BL      (BATCH * SEQLEN)                 // 8192
#define EPS     1e-5f

// offsets inside xBC row
#define XOFF_X 0
#define XOFF_B DSSM            // 2048
#define XOFF_C (DSSM + DSTATE) // 2176
// offsets inside zxbcdt row
#define ZOFF_Z  0
#define ZOFF_XBC DSSM          // 2048
#define ZOFF_DT (DSSM + CONVDIM) // 4352

// workspace offsets (in floats)
#define OFF_ZX   ((size_t)0)                         // 8192*4384
#define OFF_XBC  ((size_t)35913728)                  // 8192*2304
#define OFF_DTP  ((size_t)54788096)                  // 8192*32
#define OFF_Y    ((size_t)55050240)                  // 8192*2048
#define OFF_ST   ((size_t)71827456)                  // 2*16*32*64*128
#define OFF_CS   ((size_t)80216064)                  // 2*32*16
#define OFF_WPAD ((size_t)80217088)                  // 4480*1024
// total: 84,804,608 floats = ~339.2 MB

typedef __attribute__((ext_vector_type(2))) float v2f;
typedef __attribute__((ext_vector_type(8))) float v8f;

#define V8Z {0.f,0.f,0.f,0.f,0.f,0.f,0.f,0.f}

static __device__ __forceinline__ v8f wmma4(v2f a, v2f b, v8f c) {
  // D = A(16x4) * B(4x16) + C, all f32
  return __builtin_amdgcn_wmma_f32_16x16x4_f32(false, a, false, b, (short)0, c,
                                               false, false);
}

static __device__ __forceinline__ float siluf(float x) {
  return x / (1.f + expf(-x));
}

// ---------------------------------------------------------------------------
// Zero-pad B operand rows: dst[r,0:K) = (r < rows) ? src[r] : 0, r < rows_pad.
// ---------------------------------------------------------------------------
__global__ __launch_bounds__(256) void pad_rows(
    const float* __restrict__ src, float* __restrict__ dst, int K, int rows) {
  const int r = blockIdx.x;
  const bool valid = r < rows;
  for (int k = threadIdx.x; k < K; k += 256)
    dst[r * K + k] = valid ? src[r * K + k] : 0.f;
}

// ---------------------------------------------------------------------------
// Generic NT GEMM: C[M,N] = A[M,K] * B[N,K]^T.
// M % 128 == 0, K % 8 == 0. B must have ceil(N/128)*128 valid (padded) rows
// so fragment loads are unguarded; only the epilogue store masks col >= N.
// Workgroup tile 128x128, 8 waves of 64x32. Double-buffered fragments.
// ---------------------------------------------------------------------------
__global__ __launch_bounds__(256) void gemm_nt_f32(
    const float* __restrict__ A, const float* __restrict__ B,
    float* __restrict__ C, int M, int N, int K) {
  const int lane = threadIdx.x & 31;
  const int wid  = threadIdx.x >> 5;
  const int lrow = lane & 15;
  const int koff = (lane >> 4) << 1;
  const int hi8  = (lane >> 4) << 3;

  const int m0 = blockIdx.x * 128 + (wid & 1) * 64;
  const int n0 = blockIdx.y * 128 + (wid >> 1) * 32;

  // 32-bit element offsets -> SGPR base + VGPR offset addressing
  const int aoff = (m0 + lrow) * K + koff;
  const int boff = (n0 + lrow) * K + koff;

  v8f acc[4][2] = {{V8Z, V8Z}, {V8Z, V8Z}, {V8Z, V8Z}, {V8Z, V8Z}};

  v2f a[4], b[2];
#pragma unroll
  for (int mt = 0; mt < 4; ++mt)
    a[mt] = *(const v2f*)(A + aoff + mt * 16 * K);
#pragma unroll
  for (int nt = 0; nt < 2; ++nt)
    b[nt] = *(const v2f*)(B + boff + nt * 16 * K);

  int k0 = 0;
  for (; k0 + 4 < K; k0 += 4) {
    v2f an[4], bn[2];
#pragma unroll
    for (int mt = 0; mt < 4; ++mt)
      an[mt] = *(const v2f*)(A + aoff + mt * 16 * K + k0 + 4);
#pragma unroll
    for (int nt = 0; nt < 2; ++nt)
      bn[nt] = *(const v2f*)(B + boff + nt * 16 * K + k0 + 4);
#pragma unroll
    for (int mt = 0; mt < 4; ++mt)
#pragma unroll
      for (int nt = 0; nt < 2; ++nt)
        acc[mt][nt] = wmma4(a[mt], b[nt], acc[mt][nt]);
#pragma unroll
    for (int mt = 0; mt < 4; ++mt) a[mt] = an[mt];
#pragma unroll
    for (int nt = 0; nt < 2; ++nt) b[nt] = bn[nt];
  }
#pragma unroll
  for (int mt = 0; mt < 4; ++mt)
#pragma unroll
    for (int nt = 0; nt < 2; ++nt)
      acc[mt][nt] = wmma4(a[mt], b[nt], acc[mt][nt]);

#pragma unroll
  for (int mt = 0; mt < 4; ++mt)
#pragma unroll
    for (int nt = 0; nt < 2; ++nt)
#pragma unroll
      for (int e = 0; e < 8; ++e) {
        const int row = m0 + mt * 16 + e + hi8;
        const int col = n0 + nt * 16 + lrow;
        if (col < N) C[row * N + col] = acc[mt][nt][e];
      }
}

// ---------------------------------------------------------------------------
// Causal depthwise conv (D_CONV=4) + bias + SiLU over xBC; softplus(dt+bias).
// One workgroup per (b,l).
// ---------------------------------------------------------------------------
__global__ __launch_bounds__(256) void conv_silu_dt(
    const float* __restrict__ zx, const float* __restrict__ conv_w,
    const float* __restrict__ conv_b, const float* __restrict__ dt_bias,
    float* __restrict__ xbc, float* __restrict__ dtp) {
  const int bl = blockIdx.x;
  const int b  = bl / SEQLEN;
  const int l  = bl % SEQLEN;

  for (int c = threadIdx.x; c < CONVDIM; c += 256) {
    float acc = conv_b[c];
#pragma unroll
    for (int k = 0; k < DCONV; ++k) {
      const int ls = l - (DCONV - 1) + k;
      if (ls >= 0)
        acc += zx[(size_t)(b * SEQLEN + ls) * DINPROJ + ZOFF_XBC + c] *
               conv_w[c * DCONV + k];
    }
    xbc[(size_t)bl * CONVDIM + c] = siluf(acc);
  }
  for (int h = threadIdx.x; h < NHEADS; h += 256) {
    const float v = zx[(size_t)bl * DINPROJ + ZOFF_DT + h] + dt_bias[h];
    dtp[bl * NHEADS + h] = (v > 20.f) ? v : log1pf(expf(v));
  }
}

// ---------------------------------------------------------------------------
// Per-(b,chunk,head): cumulative decay, Y_diag = (L ∘ C·B^T)·(x·dt) + D·x,
// and chunk state = sum_j exp(csTot-cs_j) * (x·dt)_j ⊗ B_j.
// blockDim = 256 (8 waves, each owns two 16-row i-tiles).
// ---------------------------------------------------------------------------
__global__ __launch_bounds__(256) void ssd_chunk(
    const float* __restrict__ xbc, const float* __restrict__ dtp,
    const float* __restrict__ A_log, const float* __restrict__ Dp,
    float* __restrict__ y, float* __restrict__ states,
    float* __restrict__ csum) {
  const int idx = blockIdx.x;
  const int h  = idx & 31;
  const int cc = (idx >> 5) & 15;
  const int b  = idx >> 9;
  const int cbase = b * SEQLEN + cc * CHUNK;

  const int t    = threadIdx.x;
  const int lane = t & 31;
  const int wid  = t >> 5;
  const int lrow = lane & 15;
  const int koff = (lane >> 4) << 1;
  const int hi8  = (lane >> 4) << 3;

  __shared__ float s_xdt[CHUNK][HEADDIM];   // x * dt, 64 KB
  __shared__ float s_cs[CHUNK];             // inclusive cumsum(dA)
  __shared__ float s_dA[CHUNK];             // dA, later exp(csTot - cs)
  __shared__ float s_S[8][16][16];          // per-wave masked score tile

  const float Ah  = -expf(A_log[h]);
  const float dth = dtp[(cbase + t) * NHEADS + h];
  s_dA[t] = dth * Ah;
  __syncthreads();
  {
    float cs = 0.f;
    for (int j = 0; j <= t; ++j) cs += s_dA[j];
    s_cs[t] = cs;
  }
  // stage x*dt (thread-major over p for coalescing)
  for (int i = 0; i < (CHUNK * HEADDIM) / 256; ++i) {
    const int e = t + i * 256;
    const int j = e >> 6, p = e & 63;
    s_xdt[j][p] = xbc[(cbase + j) * CONVDIM + XOFF_X + h * HEADDIM + p] *
                  dtp[(cbase + j) * NHEADS + h];
  }
  __syncthreads();
  const float csTot = s_cs[CHUNK - 1];
  const float Dh = Dp[h];

  for (int itl = 0; itl < 2; ++itl) {
    const int it = wid * 2 + itl;
    const int arow = (cbase + it * 16 + lrow) * CONVDIM + XOFF_C + koff;
    v8f yacc[4] = {V8Z, V8Z, V8Z, V8Z};
    for (int jt = 0; jt <= it; ++jt) {
      const int brow = (cbase + jt * 16 + lrow) * CONVDIM + XOFF_B + koff;
      // G = C_i · B_j^T  (K = DSTATE)
      v8f g = V8Z;
      for (int n0 = 0; n0 < DSTATE; n0 += 4) {
        const v2f a  = *(const v2f*)(xbc + arow + n0);
        const v2f bb = *(const v2f*)(xbc + brow + n0);
        g = wmma4(a, bb, g);
      }
      // mask + decay, round-trip through LDS to re-layout C->A fragments
#pragma unroll
      for (int e = 0; e < 8; ++e) {
        const int il = it * 16 + e + hi8;
        const int jl = jt * 16 + lrow;
        s_S[wid][e + hi8][lrow] =
            (jl <= il) ? g[e] * expf(s_cs[il] - s_cs[jl]) : 0.f;
      }
      asm volatile("s_wait_dscnt 0" ::: "memory");
      // yacc += S · x_dt  (K = 16)
#pragma unroll
      for (int k0 = 0; k0 < 16; k0 += 4) {
        const v2f a2 = *(const v2f*)(&s_S[wid][lrow][k0 + koff]);
#pragma unroll
        for (int pt = 0; pt < 4; ++pt) {
          const int p = pt * 16 + lrow;
          v2f b2;
          b2.x = s_xdt[jt * 16 + k0 + koff][p];
          b2.y = s_xdt[jt * 16 + k0 + koff + 1][p];
          yacc[pt] = wmma4(a2, b2, yacc[pt]);
        }
      }
    }
    // + D * x, store Y (Y_off added later by ssd_yoff)
#pragma unroll
    for (int pt = 0; pt < 4; ++pt)
#pragma unroll
      for (int e = 0; e < 8; ++e) {
        const int il = it * 16 + e + hi8;
        const int gl = cbase + il;
        const int p  = pt * 16 + lrow;
        const float xv = xbc[gl * CONVDIM + XOFF_X + h * HEADDIM + p];
        y[gl * DSSM + h * HEADDIM + p] = yacc[pt][e] + Dh * xv;
      }
  }

  __syncthreads();
  s_dA[t] = expf(csTot - s_cs[t]);   // reuse as decay weights
  __syncthreads();

  // states[p,n] = sum_j (xdt[j,p]*w_j) * B_c[j,n]   (K = CHUNK)
  {
    const int pt = wid >> 1;
    const int nb = (wid & 1) * 4;
    const int p  = pt * 16 + lrow;
    v8f sacc[4] = {V8Z, V8Z, V8Z, V8Z};
    for (int k0 = 0; k0 < CHUNK; k0 += 4) {
      const int j = k0 + koff;
      v2f a;
      a.x = s_xdt[j][p] * s_dA[j];
      a.y = s_xdt[j + 1][p] * s_dA[j + 1];
      const int brow0 = (cbase + j) * CONVDIM + XOFF_B;
      const int brow1 = brow0 + CONVDIM;
#pragma unroll
      for (int nt = 0; nt < 4; ++nt) {
        const int ncol = (nb + nt) * 16 + lrow;
        v2f bb;
        bb.x = xbc[brow0 + ncol];
        bb.y = xbc[brow1 + ncol];
        sacc[nt] = wmma4(a, bb, sacc[nt]);
      }
    }
    const int stoff = (((b * NCHUNK + cc) * NHEADS) + h) * (HEADDIM * DSTATE);
#pragma unroll
    for (int nt = 0; nt < 4; ++nt)
#pragma unroll
      for (int e = 0; e < 8; ++e) {
        const int prow = pt * 16 + e + hi8;
        const int ncol = (nb + nt) * 16 + lrow;
        states[stoff + prow * DSTATE + ncol] = sacc[nt][e];
      }
  }
  if (t == 0) csum[(b * NHEADS + h) * NCHUNK + cc] = csTot;
}

// ---------------------------------------------------------------------------
// Inter-chunk state recurrence (in place): states[c] <- prev_state before c;
// carry = exp(chunkSum_c)*carry + states[c]. One workgroup per (b,h).
// ---------------------------------------------------------------------------
__global__ __launch_bounds__(256) void ssd_scan(
    float* __restrict__ states, const float* __restrict__ csum) {
  const int bh = blockIdx.x;
  const int b = bh >> 5, h = bh & 31;
  float carry[(HEADDIM * DSTATE) / 256];
#pragma unroll
  for (int i = 0; i < (HEADDIM * DSTATE) / 256; ++i) carry[i] = 0.f;
  for (int c = 0; c < NCHUNK; ++c) {
    const float e = expf(csum[(b * NHEADS + h) * NCHUNK + c]);
    const int base = (((b * NCHUNK + c) * NHEADS) + h) * (HEADDIM * DSTATE);
#pragma unroll
    for (int i = 0; i < (HEADDIM * DSTATE) / 256; ++i) {
      const int idx = base + threadIdx.x + i * 256;
      const float s = states[idx];
      states[idx] = carry[i];
      carry[i] = e * carry[i] + s;
    }
  }
}

// ---------------------------------------------------------------------------
// Y_off[i,p] += exp(cs_i) * sum_n C[i,n] * prev_state[p,n]
// ---------------------------------------------------------------------------
__global__ __launch_bounds__(256) void ssd_yoff(
    const float* __restrict__ xbc, const float* __restrict__ dtp,
    const float* __restrict__ A_log, const float* __restrict__ states,
    float* __restrict__ y) {
  const int idx = blockIdx.x;
  const int h  = idx & 31;
  const int cc = (idx >> 5) & 15;
  const int b  = idx >> 9;
  const int cbase = b * SEQLEN + cc * CHUNK;

  const int t    = threadIdx.x;
  const int lane = t & 31;
  const int wid  = t >> 5;
  const int lrow = lane & 15;
  const int koff = (lane >> 4) << 1;
  const int hi8  = (lane >> 4) << 3;

  __shared__ float s_dA[CHUNK];
  __shared__ float s_cs[CHUNK];
  const float Ah = -expf(A_log[h]);
  s_dA[t] = dtp[(cbase + t) * NHEADS + h] * Ah;
  __syncthreads();
  {
    float cs = 0.f;
    for (int j = 0; j <= t; ++j) cs += s_dA[j];
    s_cs[t] = cs;
  }
  __syncthreads();

  const int psoff = (((b * NCHUNK + cc) * NHEADS) + h) * (HEADDIM * DSTATE);
  for (int itl = 0; itl < 2; ++itl) {
    const int it = wid * 2 + itl;
    const int arow = (cbase + it * 16 + lrow) * CONVDIM + XOFF_C + koff;
    v8f yacc[4] = {V8Z, V8Z, V8Z, V8Z};
    for (int n0 = 0; n0 < DSTATE; n0 += 4) {
      const v2f a = *(const v2f*)(xbc + arow + n0);
#pragma unroll
      for (int pt = 0; pt < 4; ++pt) {
        const v2f bb =
            *(const v2f*)(states + psoff + (pt * 16 + lrow) * DSTATE + n0 + koff);
        yacc[pt] = wmma4(a, bb, yacc[pt]);
      }
    }
#pragma unroll
    for (int pt = 0; pt < 4; ++pt)
#pragma unroll
      for (int e = 0; e < 8; ++e) {
        const int il = it * 16 + e + hi8;
        const int gl = cbase + il;
        const int p  = pt * 16 + lrow;
        const int off = gl * DSSM + h * HEADDIM + p;
        y[off] += expf(s_cs[il]) * yacc[pt][e];
      }
  }
}

// ---------------------------------------------------------------------------
// y = y * silu(z); y = y * rsqrt(mean(y^2) + eps) * norm_w   (in place)
// ---------------------------------------------------------------------------
__global__ __launch_bounds__(256) void gate_rmsnorm(
    const float* __restrict__ zx, const float* __restrict__ norm_w,
    float* __restrict__ y) {
  const int bl = blockIdx.x;
  __shared__ float red[256];
  float vals[DSSM / 256];
  float ss = 0.f;
#pragma unroll
  for (int i = 0; i < DSSM / 256; ++i) {
    const int c = threadIdx.x + i * 256;
    const float zv = zx[(size_t)bl * DINPROJ + ZOFF_Z + c];
    const float yv = y[(size_t)bl * DSSM + c] * siluf(zv);
    vals[i] = yv;
    ss += yv * yv;
  }
  red[threadIdx.x] = ss;
  __syncthreads();
  for (int s = 128; s > 0; s >>= 1) {
    if (threadIdx.x < s) red[threadIdx.x] += red[threadIdx.x + s];
    __syncthreads();
  }
  const float scale = rsqrtf(red[0] / (float)DSSM + EPS);
#pragma unroll
  for (int i = 0; i < DSSM / 256; ++i) {
    const int c = threadIdx.x + i * 256;
    y[(size_t)bl * DSSM + c] = vals[i] * scale * norm_w[c];
  }
}

// ---------------------------------------------------------------------------
extern "C" void kernel_launch(void* const* d_in, const int* in_sizes, int n_in,
                              void* d_out, int out_size, void* d_ws,
                              size_t ws_size, hipStream_t stream) {
  const float* u       = (const float*)d_in[0];
  const float* W_in    = (const float*)d_in[1];
  const float* conv_w  = (const float*)d_in[2];
  const float* conv_b  = (const float*)d_in[3];
  const float* dt_bias = (const float*)d_in[4];
  const float* A_log   = (const float*)d_in[5];
  const float* Dp      = (const float*)d_in[6];
  const float* norm_w  = (const float*)d_in[7];
  const float* W_out   = (const float*)d_in[8];
  float* out = (float*)d_out;

  float* ws   = (float*)d_ws;
  float* zx   = ws + OFF_ZX;
  float* xbc  = ws + OFF_XBC;
  float* dtp  = ws + OFF_DTP;
  float* yb   = ws + OFF_Y;
  float* st   = ws + OFF_ST;
  float* cs   = ws + OFF_CS;
  float* wpad = ws + OFF_WPAD;

  const dim3 blk(256);

  // pad W_in rows 4384 -> 4480 so the GEMM inner loop needs no guards
  pad_rows<<<dim3(DINPROJ_PAD), blk, 0, stream>>>(W_in, wpad, DMODEL, DINPROJ);

  gemm_nt_f32<<<dim3(BL / 128, DINPROJ_PAD / 128), blk, 0, stream>>>(
      u, wpad, zx, BL, DINPROJ, DMODEL);

  conv_silu_dt<<<dim3(BL), blk, 0, stream>>>(zx, conv_w, conv_b, dt_bias, xbc,
                                             dtp);

  ssd_chunk<<<dim3(BATCH * NCHUNK * NHEADS), blk, 0, stream>>>(
      xbc, dtp, A_log, Dp, yb, st, cs);

  ssd_scan<<<dim3(BATCH * NHEADS), blk, 0, stream>>>(st, cs);

  ssd_yoff<<<dim3(BATCH * NCHUNK * NHEADS), blk, 0, stream>>>(xbc, dtp, A_log,
                                                              st, yb);

  gate_rmsnorm<<<dim3(BL), blk, 0, stream>>>(zx, norm_w, yb);

  gemm_nt_f32<<<dim3(BL / 128, DMODEL / 128), blk, 0, stream>>>(
      yb, W_out, out, BL, DMODEL, DINNER);
}